// GCN2_22660247453952
// MI455X (gfx1250) — compile-verified
//
#include <hip/hip_runtime.h>

#define N_NODES 100000

typedef __attribute__((ext_vector_type(2))) float v2f;
typedef __attribute__((ext_vector_type(8))) float v8f;

// ---------------- elementwise helpers ----------------

__global__ void fill_kernel(float* __restrict__ p, float v, int n) {
  int i = blockIdx.x * blockDim.x + threadIdx.x;
  if (i < n) p[i] = v;
}

__global__ void deg_count_kernel(const long long* __restrict__ col,
                                 float* __restrict__ deg, int E) {
  int e = blockIdx.x * blockDim.x + threadIdx.x;
  if (e < E) atomicAdd(&deg[(int)col[e]], 1.0f);
}

__global__ void rsqrt_kernel(float* __restrict__ p, int n) {
  int i = blockIdx.x * blockDim.x + threadIdx.x;
  if (i < n) p[i] = rsqrtf(p[i]);
}

// ---------------- WMMA f32 GEMM: H[n_rows x NT*16] = X[n_rows x K] @ W[K x wcols] ----------------
// One wave per 16-row tile. W staged in LDS pre-paired by K: layout [K/2][NT*16][2],
// so a lane's B fragment (k, k+1) is one aligned v2f -> single ds_load_b64, no shuffles.
// A-frag (16x4 f32): lane = M (0..15) twice; VGPR0/1 = K = kq, kq+1 with kq = (lane>=16)?2:0.
// B-frag (4x16 f32): lane = N twice; VGPR0/1 = K = kq, kq+1.
// C/D (16x16 f32): VGPR r -> row r (lanes 0-15) / r+8 (lanes 16-31), col = lane&15.
template <int K, int NT>
__global__ void gemm_wmma_kernel(const float* __restrict__ X,
                                 const float* __restrict__ W, int wcols,
                                 float* __restrict__ H, int n_rows) {
  __shared__ v2f sWp[(K / 2) * NT * 16];  // [kpair][n] -> (W(2p,n), W(2p+1,n))
  float* sW = (float*)sWp;
  const int tid = threadIdx.x;
  for (int idx = tid; idx < K * NT * 16; idx += 256) {
    int k = idx / (NT * 16);
    int n = idx - k * (NT * 16);
    float v = (n < wcols) ? W[k * wcols + n] : 0.0f;
    sW[((k >> 1) * (NT * 16) + n) * 2 + (k & 1)] = v;
  }
  __syncthreads();

  const int wave = tid >> 5;
  const int lane = tid & 31;
  const int tile_m = blockIdx.x * 8 + wave;
  if (tile_m * 16 >= n_rows) return;  // wave-uniform: EXEC stays all-1s below

  const int m  = lane & 15;
  const int kq = (lane >> 4) << 1;  // 0 or 2
  const v2f* xrow = (const v2f*)(X + (long long)(tile_m * 16 + m) * K);

  v8f acc[NT] = {};
  for (int kb = 0; kb < K; kb += 4) {
    const int p = (kb + kq) >> 1;         // aligned K-pair index
    v2f a = xrow[p];
#pragma unroll
    for (int t = 0; t < NT; ++t) {
      v2f b = sWp[p * (NT * 16) + t * 16 + m];
      acc[t] = __builtin_amdgcn_wmma_f32_16x16x4_f32(
          false, a, false, b, (short)0, acc[t], false, false);
    }
  }

  const int r0 = (lane >> 4) << 3;  // 0 or 8
#pragma unroll
  for (int t = 0; t < NT; ++t)
#pragma unroll
    for (int r = 0; r < 8; ++r)
      H[(long long)(tile_m * 16 + r0 + r) * (NT * 16) + t * 16 + m] = acc[t][r];
}

// ---------------- edge scatter: acc[col] += H[row] * dinv[row]*dinv[col] ----------------
template <int F>  // F multiple of 4; F/4 threads per edge, float4 gather + 4 atomics
__global__ void scatter_kernel(const long long* __restrict__ row,
                               const long long* __restrict__ col,
                               const float* __restrict__ dinv,
                               const float* __restrict__ H, int ldh,
                               float* __restrict__ acc, int lda, int E) {
  constexpr int TPE = F / 4;
  long long t = (long long)blockIdx.x * blockDim.x + threadIdx.x;
  int e = (int)(t / TPE);
  int g = (int)(t % TPE);
  if (e >= E) return;
  int r = (int)row[e];
  int c = (int)col[e];
  float w = dinv[r] * dinv[c];
  float4 hv = *(const float4*)(H + (long long)r * ldh + g * 4);
  float* dst = acc + (long long)c * lda + g * 4;
  atomicAdd(dst + 0, hv.x * w);
  atomicAdd(dst + 1, hv.y * w);
  atomicAdd(dst + 2, hv.z * w);
  atomicAdd(dst + 3, hv.w * w);
}

// ---------------- finalize: out = act(acc + H*dinv^2 (self loop) + bias) ----------------
template <bool LRELU>
__global__ void finalize_kernel(const float* __restrict__ acc, int lda,
                                const float* __restrict__ H, int ldh,
                                const float* __restrict__ dinv,
                                const float* __restrict__ bias,
                                float* __restrict__ out, int ldo, int F, int n) {
  int t = blockIdx.x * blockDim.x + threadIdx.x;
  if (t >= n * F) return;
  int i = t / F;
  int j = t - i * F;
  float di = dinv[i];
  float v = acc[(long long)i * lda + j] + H[(long long)i * ldh + j] * di * di + bias[j];
  if (LRELU) v = (v > 0.0f) ? v : 0.01f * v;
  out[(long long)i * ldo + j] = v;
}

// ---------------- driver ----------------
extern "C" void kernel_launch(void* const* d_in, const int* in_sizes, int n_in,
                              void* d_out, int out_size, void* d_ws, size_t ws_size,
                              hipStream_t stream) {
  (void)n_in; (void)out_size; (void)ws_size;
  const float*     x    = (const float*)d_in[0];
  const long long* ei   = (const long long*)d_in[1];
  const float*     W1   = (const float*)d_in[2];
  const float*     b1   = (const float*)d_in[3];
  const float*     W2   = (const float*)d_in[4];
  const float*     b2   = (const float*)d_in[5];
  const int E  = in_sizes[1] / 2;
  const int Nn = N_NODES;
  const long long* rowp = ei;      // edge_index[0] = source
  const long long* colp = ei + E;  // edge_index[1] = destination

  char* ws = (char*)d_ws;
  float* dinv = (float*)ws;                            // N floats (deg -> rsqrt in place)
  float* H1   = (float*)(ws + (size_t)(1  << 20));     // N x 64   (reused as H2: N x 48)
  float* acc1 = (float*)(ws + (size_t)(28 << 20));     // N x 64   (reused as acc2: N x 40)
  float* hid  = (float*)(ws + (size_t)(55 << 20));     // N x 64
  float* H2   = H1;
  float* acc2 = acc1;
  float* out  = (float*)d_out;

  const int B = 256;
  const int mtiles  = (Nn + 15) / 16;      // 6250
  const int gblocks = (mtiles + 7) / 8;    // 782 blocks of 8 waves

  // degrees (self loop => init 1.0) -> dinv
  fill_kernel<<<(Nn + B - 1) / B, B, 0, stream>>>(dinv, 1.0f, Nn);
  deg_count_kernel<<<(E + B - 1) / B, B, 0, stream>>>(colp, dinv, E);
  rsqrt_kernel<<<(Nn + B - 1) / B, B, 0, stream>>>(dinv, Nn);

  // layer 1: H1 = x @ W1 ; scatter ; hid = leaky_relu(acc + self + b1)
  gemm_wmma_kernel<128, 4><<<gblocks, 256, 0, stream>>>(x, W1, 64, H1, Nn);
  fill_kernel<<<(Nn * 64 + B - 1) / B, B, 0, stream>>>(acc1, 0.0f, Nn * 64);
  {
    long long tt = (long long)E * 16;
    scatter_kernel<64><<<(int)((tt + B - 1) / B), B, 0, stream>>>(
        rowp, colp, dinv, H1, 64, acc1, 64, E);
  }
  finalize_kernel<true><<<(Nn * 64 + B - 1) / B, B, 0, stream>>>(
      acc1, 64, H1, 64, dinv, b1, hid, 64, 64, Nn);

  // layer 2: H2 = hid @ W2 (40 cols padded to 48) ; scatter ; out = acc + self + b2
  gemm_wmma_kernel<64, 3><<<gblocks, 256, 0, stream>>>(hid, W2, 40, H2, Nn);
  fill_kernel<<<(Nn * 40 + B - 1) / B, B, 0, stream>>>(acc2, 0.0f, Nn * 40);
  {
    long long tt = (long long)E * 10;
    scatter_kernel<40><<<(int)((tt + B - 1) / B), B, 0, stream>>>(
        rowp, colp, dinv, H2, 48, acc2, 40, E);
  }
  finalize_kernel<false><<<(Nn * 40 + B - 1) / B, B, 0, stream>>>(
      acc2, 40, H2, 48, dinv, b2, out, 40, 40, Nn);
}